// SquareDiagonalLinear_71674414235906
// MI455X (gfx1250) — compile-verified
//
#include <hip/hip_runtime.h>

// out[b,n] = input[b,n] * weight[0] + bias[n]
// B=4096, N=8192, fp32. Pure streaming op: ~268 MB moved, ~67 MFLOP.
// Bound by HBM (23.3 TB/s -> ~11.5 us floor). Optimal path: coalesced B128
// non-temporal loads/stores, 4 outstanding loads per thread, bias cached.

typedef __attribute__((ext_vector_type(4))) float f32x4;

__global__ __launch_bounds__(256) void scale_bias_stream_kernel(
    const f32x4* __restrict__ in4,
    const float* __restrict__ weight,
    const f32x4* __restrict__ bias4,
    f32x4* __restrict__ out4,
    long long total4,      // total float4 elements = B*N/4
    unsigned n4mask)       // (N/4) - 1, N/4 is a power of two
{
    const float w = weight[0];  // uniform; lands in cache, broadcast to wave

    // Block covers blockDim.x*4 consecutive float4s; each of the 4 unrolled
    // steps is a fully coalesced wave-wide B128 access (512 B contiguous/wave).
    const long long base =
        (long long)blockIdx.x * ((long long)blockDim.x * 4) + threadIdx.x;

#pragma unroll
    for (int j = 0; j < 4; ++j) {
        const long long idx4 = base + (long long)j * blockDim.x;
        if (idx4 < total4) {
            // Input is streamed exactly once -> non-temporal load.
            f32x4 x = __builtin_nontemporal_load(&in4[idx4]);
            // Bias: 32 KB, reused by all 4096 rows -> regular (temporal) load.
            f32x4 b = bias4[(unsigned)idx4 & n4mask];
            f32x4 y = x * w + b;  // vector FMA
            // Output is written exactly once -> non-temporal store.
            __builtin_nontemporal_store(y, &out4[idx4]);
        }
    }
}

extern "C" void kernel_launch(void* const* d_in, const int* in_sizes, int n_in,
                              void* d_out, int out_size, void* d_ws, size_t ws_size,
                              hipStream_t stream) {
    (void)d_ws; (void)ws_size; (void)n_in; (void)out_size;

    const f32x4* in4   = (const f32x4*)d_in[0];   // input  (B*N fp32)
    const float* wgt   = (const float*)d_in[1];   // weight (1 fp32)
    const f32x4* bias4 = (const f32x4*)d_in[2];   // bias   (N fp32)
    f32x4*       out4  = (f32x4*)d_out;

    const long long total  = (long long)in_sizes[0];      // B*N
    const long long total4 = total >> 2;                  // float4 count
    const unsigned  n4     = (unsigned)(in_sizes[2] >> 2); // N/4 (power of two)
    const unsigned  n4mask = n4 - 1u;

    const int  threads   = 256;
    const long long perBlock = (long long)threads * 4;
    const int  blocks    = (int)((total4 + perBlock - 1) / perBlock);

    scale_bias_stream_kernel<<<blocks, threads, 0, stream>>>(
        in4, wgt, bias4, out4, total4, n4mask);
}